// Transformer_91199335563579
// MI455X (gfx1250) — compile-verified
//
#include <hip/hip_runtime.h>
#include <math.h>

// ---------------------------------------------------------------------------
// Model constants (match reference)
// ---------------------------------------------------------------------------
#define D_MODEL   4096
#define N_HEADS   32
#define N_KV      8
#define HEAD_DIM  128
#define HIDDEN    14336
#define VOCAB     32000
#define MAX_SEQ   2048
#define BATCH     8

typedef float v2f __attribute__((ext_vector_type(2)));
typedef float v8f __attribute__((ext_vector_type(8)));

// ---------------------------------------------------------------------------
// Workspace layout (floats). All GEMM "A" buffers are padded to 16 rows so the
// 16x16x4 WMMA tile has defined (zero) data in rows 8..15.
// ---------------------------------------------------------------------------
#define OFF_H     0                        // 16 x 4096  residual stream
#define OFF_XN    (OFF_H   + 16*D_MODEL)   // 16 x 4096  rmsnorm output
#define OFF_Q     (OFF_XN  + 16*D_MODEL)   // 16 x 4096  xq (after rope)
#define OFF_KX    (OFF_Q   + 16*D_MODEL)   // 16 x 1024  new k (after rope)
#define OFF_VX    (OFF_KX  + 16*1024)      // 16 x 1024  new v
#define OFF_AT    (OFF_VX  + 16*1024)      // 16 x 4096  attention output
#define OFF_G1    (OFF_AT  + 16*D_MODEL)   // 16 x 14336 hn @ w1
#define OFF_U3    (OFF_G1  + 16*HIDDEN)    // 16 x 14336 hn @ w3
#define OFF_FF    (OFF_U3  + 16*HIDDEN)    // 16 x 14336 silu(g)*u
#define WS_FLOATS (OFF_FF  + 16*HIDDEN)    // ~0.98M floats (~3.9 MB)

// ---------------------------------------------------------------------------
// Zero-fill (workspace is poisoned by harness; pad rows must be zero)
// ---------------------------------------------------------------------------
__global__ void k_zero(float* __restrict__ p, long n) {
    long i = (long)blockIdx.x * blockDim.x + threadIdx.x;
    if (i < n) p[i] = 0.0f;
}

// ---------------------------------------------------------------------------
// Embedding gather + RMSNorm(gamma1). One block per batch row.
// ---------------------------------------------------------------------------
__global__ __launch_bounds__(256) void k_embed_norm(
    const int* __restrict__ tokens, const float* __restrict__ emb,
    const float* __restrict__ gamma, float* __restrict__ h, float* __restrict__ xn)
{
    const int b = blockIdx.x;
    const int tok = tokens[b];
    const float* src = emb + (long)tok * D_MODEL;
    __shared__ float red[256];
    float ss = 0.0f;
    for (int i = threadIdx.x; i < D_MODEL; i += 256) {
        float v = src[i];
        h[b * D_MODEL + i] = v;
        ss += v * v;
    }
    red[threadIdx.x] = ss; __syncthreads();
    for (int s = 128; s > 0; s >>= 1) {
        if (threadIdx.x < s) red[threadIdx.x] += red[threadIdx.x + s];
        __syncthreads();
    }
    const float rs = rsqrtf(red[0] / (float)D_MODEL);
    for (int i = threadIdx.x; i < D_MODEL; i += 256)
        xn[b * D_MODEL + i] = h[b * D_MODEL + i] * rs * gamma[i];
}

// RMSNorm of an existing 16x4096 buffer (rows < 8 valid)
__global__ __launch_bounds__(256) void k_rmsnorm(
    const float* __restrict__ X, const float* __restrict__ gamma, float* __restrict__ Out)
{
    const int b = blockIdx.x;
    __shared__ float red[256];
    float ss = 0.0f;
    for (int i = threadIdx.x; i < D_MODEL; i += 256) {
        float v = X[b * D_MODEL + i];
        ss += v * v;
    }
    red[threadIdx.x] = ss; __syncthreads();
    for (int s = 128; s > 0; s >>= 1) {
        if (threadIdx.x < s) red[threadIdx.x] += red[threadIdx.x + s];
        __syncthreads();
    }
    const float rs = rsqrtf(red[0] / (float)D_MODEL);
    for (int i = threadIdx.x; i < D_MODEL; i += 256)
        Out[b * D_MODEL + i] = X[b * D_MODEL + i] * rs * gamma[i];
}

// ---------------------------------------------------------------------------
// WMMA fp32 GEMM: Out[16 x N] = A[16 x K] * W[K x N] (+ R).
// One wave (32 threads) per 16-column tile; V_WMMA_F32_16X16X4_F32, K step 4.
// A-operand layout (16x4 f32): lanes 0-15 -> K pair {k,k+1}, lanes 16-31 ->
// {k+2,k+3}; row m = lane%16. B mirrored: col n = n0 + lane%16.
// C/D layout: VGPR r, lane l -> row r + 8*(l/16), col l%16.
// A rows 8..15 are zero, so pad output rows come out zero.
// ---------------------------------------------------------------------------
__global__ __launch_bounds__(32) void k_gemm_wmma(
    const float* __restrict__ A,   // 16 x K (rows 8..15 zero)
    const float* __restrict__ W,   // K x N row-major
    const float* __restrict__ R,   // residual 16 x N or nullptr
    float* __restrict__ Out,       // out_rows x N
    int K, int N, int out_rows)
{
    const int lane = threadIdx.x & 31;
    const int half = lane >> 4;        // 0 -> K {0,1}; 1 -> K {2,3}
    const int li   = lane & 15;
    const int n0   = blockIdx.x << 4;
    if (n0 >= N) return;
    const int n = n0 + li;

    const float* aRow = A + (long)li * K + 2 * half;
    const float* wCol = W + (long)(2 * half) * N + n;

    v8f acc = {0.f, 0.f, 0.f, 0.f, 0.f, 0.f, 0.f, 0.f};

    for (int k = 0; k < K; k += 4) {
        if ((k & 63) == 0) {
            // pull the weight stream ahead into cache (global_prefetch)
            __builtin_prefetch(wCol + (long)(k + 128) * N, 0, 0);
        }
        v2f a, b;
        a.x = aRow[k];
        a.y = aRow[k + 1];
        b.x = wCol[(long)k * N];
        b.y = wCol[(long)(k + 1) * N];
        acc = __builtin_amdgcn_wmma_f32_16x16x4_f32(
            false, a, false, b, (short)0, acc, false, false);
    }

    #pragma unroll
    for (int r = 0; r < 8; ++r) {
        const int m = r + 8 * half;
        float val = acc[r];
        if (R) val += R[(long)m * N + n];      // R pad rows are zero
        if (m < out_rows) Out[(long)m * N + n] = val;
    }
}

// ---------------------------------------------------------------------------
// RoPE, applied in place. Pairs are interleaved (even, odd) per reference.
// ---------------------------------------------------------------------------
__global__ void k_rope(float* __restrict__ X, const int* __restrict__ start_pos,
                       int n_heads, int row_stride)
{
    const int idx = blockIdx.x * blockDim.x + threadIdx.x;
    const int total = BATCH * n_heads * (HEAD_DIM / 2);
    if (idx >= total) return;
    const int i = idx & 63;                    // pair index 0..63
    const int h = (idx >> 6) % n_heads;
    const int b = idx / (n_heads * 64);
    float* p = X + (long)b * row_stride + h * HEAD_DIM + 2 * i;
    const float x0 = p[0], x1 = p[1];
    const float freq = __expf(-(float)(2 * i) * (logf(10000.0f) / (float)HEAD_DIM));
    const float ang  = (float)(*start_pos) * freq;
    float s, c;
    __sincosf(ang, &s, &c);
    p[0] = x0 * c - x1 * s;
    p[1] = x0 * s + x1 * c;
}

// ---------------------------------------------------------------------------
// GQA decode attention. One block per (kv_head, batch); 4 query heads share
// the KV stream. Key j == start_pos comes from the fresh K/V (functional
// cache update -- inputs are never mutated). Scores + softmax live in LDS
// (4 heads x 2048 x f32 = 32 KB of the 320 KB WGP LDS).
// ---------------------------------------------------------------------------
__global__ __launch_bounds__(256) void k_attn(
    const float* __restrict__ Q,       // 16 x 4096 (rows < 8 valid)
    const float* __restrict__ Knew,    // 16 x 1024
    const float* __restrict__ Vnew,    // 16 x 1024
    const float* __restrict__ cacheK,  // [8][2048][8][128]
    const float* __restrict__ cacheV,
    const int* __restrict__ start_pos_p,
    float* __restrict__ Out)           // 16 x 4096
{
    const int kv   = blockIdx.x;       // 0..7
    const int b    = blockIdx.y;       // 0..7
    const int tid  = threadIdx.x;
    const int lane = tid & 31;
    const int wave = tid >> 5;
    const int sp   = *start_pos_p;
    const int L    = sp + 1;           // <= 2048

    __shared__ float sc[4][MAX_SEQ];
    __shared__ float red[256];

    // Per-lane slice of q for the 4 heads sharing this kv head
    const int d0 = lane * 4;
    float4 qr[4];
    #pragma unroll
    for (int hh = 0; hh < 4; ++hh) {
        const float* qp = Q + (long)b * D_MODEL + (kv * 4 + hh) * HEAD_DIM + d0;
        qr[hh] = make_float4(qp[0], qp[1], qp[2], qp[3]);
    }
    const float scale = 0.08838834764831845f; // 1/sqrt(128)

    // Phase A: scores.  Wave w handles keys w, w+8, ... Each wave loads a full
    // coalesced 512B key row (4 floats/lane), wave32 butterfly reduce.
    for (int j = wave; j < L; j += 8) {
        const float* kp = (j < sp)
            ? cacheK + ((((long)b * MAX_SEQ + j) * N_KV + kv) * HEAD_DIM) + d0
            : Knew + (long)b * (N_KV * HEAD_DIM) + kv * HEAD_DIM + d0;
        const float4 kk = make_float4(kp[0], kp[1], kp[2], kp[3]);
        #pragma unroll
        for (int hh = 0; hh < 4; ++hh) {
            float p = qr[hh].x * kk.x + qr[hh].y * kk.y
                    + qr[hh].z * kk.z + qr[hh].w * kk.w;
            p += __shfl_xor(p, 16);
            p += __shfl_xor(p, 8);
            p += __shfl_xor(p, 4);
            p += __shfl_xor(p, 2);
            p += __shfl_xor(p, 1);
            if (lane == 0) sc[hh][j] = p * scale;
        }
    }
    __syncthreads();

    // Phase B: softmax per head
    for (int hh = 0; hh < 4; ++hh) {
        float lm = -3.4e38f;
        for (int j = tid; j < L; j += 256) lm = fmaxf(lm, sc[hh][j]);
        red[tid] = lm; __syncthreads();
        for (int s = 128; s > 0; s >>= 1) {
            if (tid < s) red[tid] = fmaxf(red[tid], red[tid + s]);
            __syncthreads();
        }
        const float mx = red[0]; __syncthreads();

        float ls = 0.0f;
        for (int j = tid; j < L; j += 256) {
            const float e = __expf(sc[hh][j] - mx);
            sc[hh][j] = e;
            ls += e;
        }
        red[tid] = ls; __syncthreads();
        for (int s = 128; s > 0; s >>= 1) {
            if (tid < s) red[tid] += red[tid + s];
            __syncthreads();
        }
        const float inv = 1.0f / red[0]; __syncthreads();
        for (int j = tid; j < L; j += 256) sc[hh][j] *= inv;
        __syncthreads();
    }

    // Phase C: weighted V sum. tid -> (d, head-pair); V loads coalesced in d.
    const int d  = tid & 127;
    const int hp = tid >> 7;           // 0 or 1
    const int h0 = hp * 2, h1 = hp * 2 + 1;
    float acc0 = 0.0f, acc1 = 0.0f;
    for (int j = 0; j < L; ++j) {
        const float v = (j < sp)
            ? cacheV[((((long)b * MAX_SEQ + j) * N_KV + kv) * HEAD_DIM) + d]
            : Vnew[(long)b * (N_KV * HEAD_DIM) + kv * HEAD_DIM + d];
        acc0 += sc[h0][j] * v;
        acc1 += sc[h1][j] * v;
    }
    Out[(long)b * D_MODEL + (kv * 4 + h0) * HEAD_DIM + d] = acc0;
    Out[(long)b * D_MODEL + (kv * 4 + h1) * HEAD_DIM + d] = acc1;
}

// ---------------------------------------------------------------------------
// SwiGLU elementwise: ff = silu(g) * u  (rows < 8, contiguous in padded buf)
// ---------------------------------------------------------------------------
__global__ void k_silu_mul(const float* __restrict__ G, const float* __restrict__ U,
                           float* __restrict__ F)
{
    const int idx = blockIdx.x * blockDim.x + threadIdx.x;
    if (idx >= BATCH * HIDDEN) return;
    const float g = G[idx];
    const float u = U[idx];
    F[idx] = (g / (1.0f + __expf(-g))) * u;
}

// ---------------------------------------------------------------------------
// kernel_launch
// ---------------------------------------------------------------------------
extern "C" void kernel_launch(void* const* d_in, const int* in_sizes, int n_in,
                              void* d_out, int out_size, void* d_ws, size_t ws_size,
                              hipStream_t stream)
{
    (void)in_sizes; (void)n_in; (void)out_size; (void)ws_size;

    const int*   tokens    = (const int*)  d_in[0];
    const int*   start_pos = (const int*)  d_in[1];
    const float* tok_emb   = (const float*)d_in[2];
    const float* gamma1    = (const float*)d_in[3];
    const float* gamma2    = (const float*)d_in[4];
    const float* gamma_f   = (const float*)d_in[5];
    const float* wq        = (const float*)d_in[6];
    const float* wk        = (const float*)d_in[7];
    const float* wv        = (const float*)d_in[8];
    const float* wo        = (const float*)d_in[9];
    const float* w1        = (const float*)d_in[10];
    const float* w2        = (const float*)d_in[11];
    const float* w3        = (const float*)d_in[12];
    const float* w_out     = (const float*)d_in[13];
    const float* cache_k   = (const float*)d_in[14];
    const float* cache_v   = (const float*)d_in[15];
    float*       out       = (float*)d_out;

    float* ws = (float*)d_ws;
    float* h    = ws + OFF_H;
    float* xn   = ws + OFF_XN;
    float* q    = ws + OFF_Q;
    float* kx   = ws + OFF_KX;
    float* vx   = ws + OFF_VX;
    float* atn  = ws + OFF_AT;
    float* g1   = ws + OFF_G1;
    float* u3   = ws + OFF_U3;
    float* ff   = ws + OFF_FF;

    // 0) zero workspace (pad rows of all WMMA A/out buffers must be 0)
    k_zero<<<(WS_FLOATS + 255) / 256, 256, 0, stream>>>(ws, (long)WS_FLOATS);

    // 1) embedding gather + rmsnorm(gamma1)
    k_embed_norm<<<BATCH, 256, 0, stream>>>(tokens, tok_emb, gamma1, h, xn);

    // 2) QKV projections (WMMA fp32)
    k_gemm_wmma<<<D_MODEL / 16, 32, 0, stream>>>(xn, wq, nullptr, q,  D_MODEL, D_MODEL, 16);
    k_gemm_wmma<<<1024 / 16,    32, 0, stream>>>(xn, wk, nullptr, kx, D_MODEL, 1024,    16);
    k_gemm_wmma<<<1024 / 16,    32, 0, stream>>>(xn, wv, nullptr, vx, D_MODEL, 1024,    16);

    // 3) RoPE on q and new k
    k_rope<<<(BATCH * N_HEADS * 64 + 255) / 256, 256, 0, stream>>>(q,  start_pos, N_HEADS, D_MODEL);
    k_rope<<<(BATCH * N_KV    * 64 + 255) / 256, 256, 0, stream>>>(kx, start_pos, N_KV,    1024);

    // 4) attention over cache + fresh kv
    dim3 ag(N_KV, BATCH);
    k_attn<<<ag, 256, 0, stream>>>(q, kx, vx, cache_k, cache_v, start_pos, atn);

    // 5) output projection + residual (h = h + atn @ wo)
    k_gemm_wmma<<<D_MODEL / 16, 32, 0, stream>>>(atn, wo, h, h, D_MODEL, D_MODEL, 16);

    // 6) rmsnorm(gamma2) -> FFN
    k_rmsnorm<<<BATCH, 256, 0, stream>>>(h, gamma2, xn);
    k_gemm_wmma<<<HIDDEN / 16, 32, 0, stream>>>(xn, w1, nullptr, g1, D_MODEL, HIDDEN, 16);
    k_gemm_wmma<<<HIDDEN / 16, 32, 0, stream>>>(xn, w3, nullptr, u3, D_MODEL, HIDDEN, 16);
    k_silu_mul<<<(BATCH * HIDDEN + 255) / 256, 256, 0, stream>>>(g1, u3, ff);
    k_gemm_wmma<<<D_MODEL / 16, 32, 0, stream>>>(ff, w2, h, h, HIDDEN, D_MODEL, 16);

    // 7) final norm + vocab projection straight into d_out (8 x 32000)
    k_rmsnorm<<<BATCH, 256, 0, stream>>>(h, gamma_f, xn);
    k_gemm_wmma<<<VOCAB / 16, 32, 0, stream>>>(xn, w_out, nullptr, out, D_MODEL, VOCAB, 8);
}